// Model_20624432955660
// MI455X (gfx1250) — compile-verified
//
#include <hip/hip_runtime.h>
#include <hip/hip_bf16.h>

#define NUM_ENT 200000
#define ALPHA 0.2f
#define NEG_INF -9e15f

typedef __attribute__((ext_vector_type(2))) float v2f;
typedef __attribute__((ext_vector_type(8))) float v8f;

// ---------------------------------------------------------------------------
// Kernel 1: precompute Wr[64,128] = rel_table(zero-padded to 64 rows) @ fc_W^T
// via V_WMMA_F32_16X16X4_F32, and bdotr[q] = fc_b . rel_table[q].
//   Wr[q,k] = sum_d fc_W[k,d] * rel_table[q,d]  (q<51 valid; rows 51..63 junk)
// 128 threads = 4 waves; wave w owns M-tile w; 8 N-tiles x 16 K-steps each.
// A and B are staged in LDS (branch-free ds_load_b64 feed); A fragments are
// hoisted out of the N-tile loop (invariant across N).
// ---------------------------------------------------------------------------
__global__ void __launch_bounds__(128)
kgat_precompute_wr(const float* __restrict__ rel_table, // [51,64]
                   const float* __restrict__ fc_W,      // [128,64]
                   const float* __restrict__ fc_b,      // [64]
                   float* __restrict__ Wr,              // [64,128]
                   float* __restrict__ bdotr)           // [64]
{
    __shared__ __align__(16) float relS[64 * 64];   // 16 KB, zero padded
    __shared__ __align__(16) float wS[128 * 64];    // 32 KB

    // stage rel_table (pad rows 51..63 with zeros) and fc_W into LDS
    for (int i = threadIdx.x; i < 64 * 64; i += 128) {
        const int r = i >> 6;
        relS[i] = (r < 51) ? rel_table[i] : 0.0f;
    }
    for (int i = threadIdx.x; i < 128 * 64; i += 128)
        wS[i] = fc_W[i];
    __syncthreads();

    const int lane = threadIdx.x & 31;
    const int wave = threadIdx.x >> 5;      // M-tile 0..3
    const int mrow = lane & 15;             // A row-in-tile / B,D column
    const int kgrp = lane >> 4;             // 0: K={0,1}, 1: K={2,3}

    // preload all 16 A fragments for this wave's M-tile (reused by all N-tiles)
    const float* arow_p = &relS[(wave * 16 + mrow) * 64];
    v2f afrag[16];
#pragma unroll
    for (int ks = 0; ks < 16; ++ks) {
        const int d0 = ks * 4 + kgrp * 2;
        afrag[ks].x = arow_p[d0];
        afrag[ks].y = arow_p[d0 + 1];
    }

    for (int nt = 0; nt < 8; ++nt) {
        const float* bcol = &wS[(nt * 16 + mrow) * 64];  // fc_W row j = B column
        v8f c = {};
#pragma unroll
        for (int ks = 0; ks < 16; ++ks) {
            const int d0 = ks * 4 + kgrp * 2;
            v2f b;
            b.x = bcol[d0];          // B[d0][j]   = W[j][d0]
            b.y = bcol[d0 + 1];      // B[d0+1][j]
            c = __builtin_amdgcn_wmma_f32_16x16x4_f32(
                    false, afrag[ks], false, b, (short)0, c, false, false);
        }
        // D layout: VGPR v, lanes 0-15 -> M=v, lanes 16-31 -> M=v+8; N=lane&15
        const int mbase = wave * 16 + kgrp * 8;
        const int ncol  = nt * 16 + mrow;
#pragma unroll
        for (int v = 0; v < 8; ++v)
            Wr[(mbase + v) * 128 + ncol] = c[v];
    }

    // bdotr (fc_b is zero in setup, but stay faithful)
    if (threadIdx.x < 64) {
        float s = 0.0f;
        if (threadIdx.x < 51) {
            for (int d = 0; d < 64; ++d)
                s += fc_b[d] * relS[threadIdx.x * 64 + d];
        }
        bdotr[threadIdx.x] = s;
    }
}

// ---------------------------------------------------------------------------
// Kernel 2: main fused gather + attention + aggregate.
// 256 threads = 8 waves, one item per wave.
// LDS: per-wave 16x64 entity tile (4KB) + 64-float item embedding.
// ---------------------------------------------------------------------------
__global__ void __launch_bounds__(256)
kgat_attention(const int*   __restrict__ item_ids,       // [N]
               const int*   __restrict__ item_entities,  // [N,16]
               const int*   __restrict__ item_relations, // [N,16]
               const float* __restrict__ item_table,     // [N,64]
               const float* __restrict__ entity_table,   // [NUM_ENT+1,64]
               const float* __restrict__ Wr,             // [64,128]
               const float* __restrict__ bdotr,          // [64]
               float*       __restrict__ out,            // [N,64]
               int N)
{
    __shared__ __align__(16) float ebuf[8][16 * 64];  // 32 KB
    __shared__ __align__(16) float hbuf[8][64];       //  2 KB

    const int lane = threadIdx.x & 31;
    const int w    = threadIdx.x >> 5;
    const int n    = blockIdx.x * 8 + w;
    const bool ok  = (n < N);

    float h0 = 0.0f, h1 = 0.0f;

    if (ok) {
        // ---- stage item embedding ----
        const int iid = item_ids[n];
        h0 = item_table[iid * 64 + lane];
        h1 = item_table[iid * 64 + 32 + lane];
        hbuf[w][lane]      = h0;
        hbuf[w][lane + 32] = h1;

        // ---- stage 16x64 entity tile (coalesced float4) ----
        const float4* etab4 = reinterpret_cast<const float4*>(entity_table);
        float4* eb4 = reinterpret_cast<float4*>(&ebuf[w][0]);
#pragma unroll
        for (int it = 0; it < 8; ++it) {
            const int j   = it * 32 + lane;     // 0..255 float4 slots
            const int row = j >> 4;             // neighbor m
            const int c4  = j & 15;
            const int eid = item_entities[n * 16 + row];
            eb4[row * 16 + c4] = etab4[(size_t)eid * 16 + c4];
        }
    }
    __syncthreads();

    if (!ok) return;

    const float4* eb4 = reinterpret_cast<const float4*>(&ebuf[w][0]);
    const float4* hb4 = reinterpret_cast<const float4*>(&hbuf[w][0]);

    // ---- phase 1: e_m = h.Wr1[q] + ent_m.Wr2[q] + bdotr[q] ----
    const int m    = lane >> 1;
    const int half = lane & 1;
    const int q    = item_relations[n * 16 + m];

    const float4* wr4 = reinterpret_cast<const float4*>(Wr + q * 128);
    const float4* my4 = eb4 + m * 16 + half * 8;

    float acc = 0.0f;
#pragma unroll
    for (int i = 0; i < 8; ++i) {
        const float4 hv = hb4[half * 8 + i];
        const float4 w1 = wr4[half * 8 + i];
        const float4 ev = my4[i];
        const float4 w2 = wr4[16 + half * 8 + i];
        acc += hv.x * w1.x + hv.y * w1.y + hv.z * w1.z + hv.w * w1.w;
        acc += ev.x * w2.x + ev.y * w2.y + ev.z * w2.z + ev.w * w2.w;
    }
    acc += __shfl_xor(acc, 1);          // combine the two halves
    acc += bdotr[q];

    float e = (acc > 0.0f) ? acc : ALPHA * acc;        // leaky_relu
    const int eidm = item_entities[n * 16 + m];
    const float logit = (eidm != NUM_ENT) ? e : NEG_INF;

    // softmax over 16 m's (each value duplicated in a lane pair)
    float mx = logit;
#pragma unroll
    for (int d = 1; d < 32; d <<= 1) mx = fmaxf(mx, __shfl_xor(mx, d));
    const float ex = __expf(logit - mx);
    float s = ex;
#pragma unroll
    for (int d = 1; d < 32; d <<= 1) s += __shfl_xor(s, d);
    const float att = ex * (2.0f / s);  // s counts every m twice

    // ---- phase 2: out = sum_m att_m * ent_m + h ----
    float acc0 = h0, acc1 = h1;
    const float* eb = &ebuf[w][0];
#pragma unroll
    for (int mm = 0; mm < 16; ++mm) {
        const float a = __shfl(att, 2 * mm);
        acc0 += a * eb[mm * 64 + lane];
        acc1 += a * eb[mm * 64 + 32 + lane];
    }
    out[(size_t)n * 64 + lane]      = acc0;
    out[(size_t)n * 64 + 32 + lane] = acc1;
}

// ---------------------------------------------------------------------------
extern "C" void kernel_launch(void* const* d_in, const int* in_sizes, int n_in,
                              void* d_out, int out_size, void* d_ws, size_t ws_size,
                              hipStream_t stream) {
    const int*   item_ids       = (const int*)  d_in[0];
    const int*   item_entities  = (const int*)  d_in[1];
    const int*   item_relations = (const int*)  d_in[2];
    const float* item_table     = (const float*)d_in[3];
    const float* entity_table   = (const float*)d_in[4];
    const float* relation_table = (const float*)d_in[5];
    const float* fc_W           = (const float*)d_in[6];
    const float* fc_b           = (const float*)d_in[7];
    float*       out            = (float*)d_out;

    const int N = in_sizes[0];

    float* Wr    = (float*)d_ws;          // 64*128 floats = 32 KB
    float* bdotr = Wr + 64 * 128;         // 64 floats

    kgat_precompute_wr<<<1, 128, 0, stream>>>(relation_table, fc_W, fc_b, Wr, bdotr);

    const int blocks = (N + 7) / 8;
    kgat_attention<<<blocks, 256, 0, stream>>>(item_ids, item_entities,
                                               item_relations, item_table,
                                               entity_table, Wr, bdotr, out, N);
}